// TrackingHead_17386027614657
// MI455X (gfx1250) — compile-verified
//
#include <hip/hip_runtime.h>

typedef float v2f __attribute__((ext_vector_type(2)));
typedef float v4f __attribute__((ext_vector_type(4)));
typedef float v8f __attribute__((ext_vector_type(8)));

#define BN_   8
#define CN_   256
#define HWN_  65536
#define ON_   16
#define EN_   256
#define NIN_  512
#define NSLAB 64
#define SLAB  (HWN_ / NSLAB)      // 1024 hw per block
#define SUBS  (SLAB / 4)          // 256 hw per wave
#define INV_HW (1.0f / 65536.0f)

__device__ __forceinline__ v8f wmma4(v2f a, v2f b, v8f c) {
    // D = A(16x4 f32) x B(4x16 f32) + C(16x16 f32)
    return __builtin_amdgcn_wmma_f32_16x16x4_f32(
        false, a, false, b, (short)0, c, false, false);
}

__device__ __forceinline__ v8f zero8() {
    v8f z = {0.f, 0.f, 0.f, 0.f, 0.f, 0.f, 0.f, 0.f};
    return z;
}

// ---------------------------------------------------------------------------
// Kernel 0: zero the fv accumulator [B*O, C] in workspace.
// ---------------------------------------------------------------------------
__global__ __launch_bounds__(256) void zero_kernel(float* p, int n) {
    int i = blockIdx.x * 256 + threadIdx.x;
    if (i < n) p[i] = 0.f;
}

// ---------------------------------------------------------------------------
// Kernel 1: masked mean pooling as WMMA GEMM  fv[b,o,c] = sum_hw m[b,o,hw]*f[b,c,hw]
// grid = (NSLAB, B), block = 256 (8 waves).
// wave: half = w&1 selects channels [half*128, half*128+128); sub = w>>1 selects
// a 256-hw sub-slab. 8 channel tiles of 16 -> 8 WMMA accumulators per wave.
//
// Per 8-hw super-chunk each lane does one contiguous b128 NT load at
// hw + (lane>>4)*4:  .xy feeds WMMA#0 (hw {0,1,4,5} as K0..K3),
//                    .zw feeds WMMA#1 (hw {2,3,6,7} as K0..K3).
// Same permutation for A (masks) and B (features) => products pair correctly
// under the commutative hw-sum.
// ---------------------------------------------------------------------------
__global__ __launch_bounds__(256) void pool_kernel(const float* __restrict__ feat,
                                                   const float* __restrict__ mask,
                                                   float* __restrict__ fvacc) {
    __shared__ float red[2 * 8 * 256];   // [half][tile][16o x 16c] = 16 KB

    const int tid  = threadIdx.x;
    const int lane = tid & 31;
    const int w    = tid >> 5;
    const int b    = blockIdx.y;
    const int half = w & 1;
    const int sub  = w >> 1;
    const int l15  = lane & 15;
    const int hi4  = (lane >> 4) << 2;          // 0 for lanes 0-15, 4 for 16-31
    const int hw0  = blockIdx.x * SLAB + sub * SUBS;

    for (int i = tid; i < 2 * 8 * 256; i += 256) red[i] = 0.f;
    __syncthreads();

    v8f acc[8];
#pragma unroll
    for (int t = 0; t < 8; ++t) acc[t] = zero8();

    // A rows: masks[b, o=l15, hw];  B cols: features[b, c=half*128+t*16+l15, hw]
    const float* mrow  = mask + ((size_t)b * ON_ + l15) * HWN_;
    const float* fbase = feat + ((size_t)b * CN_ + (half * 128 + l15)) * HWN_;

    for (int k = 0; k < SUBS; k += 8) {
        const int hw = hw0 + k + hi4;
        v4f am = __builtin_nontemporal_load((const v4f*)(mrow + hw));
        const v2f a0 = am.xy;
        const v2f a1 = am.zw;
#pragma unroll
        for (int t = 0; t < 8; ++t) {
            v4f bm = __builtin_nontemporal_load(
                (const v4f*)(fbase + (size_t)t * 16 * HWN_ + hw));
            acc[t] = wmma4(a0, bm.xy, acc[t]);
            acc[t] = wmma4(a1, bm.zw, acc[t]);
        }
    }

    // Cross-wave (4 sub-slabs) reduce in LDS with ds_add_f32.
#pragma unroll
    for (int t = 0; t < 8; ++t) {
#pragma unroll
        for (int j = 0; j < 8; ++j) {
            const int r = (lane < 16) ? j : (j + 8);         // D-matrix row
            atomicAdd(&red[(half * 8 + t) * 256 + r * 16 + l15], acc[t][j]);
        }
    }
    __syncthreads();

    // One wave per half flushes to global accumulator with atomic f32 adds.
    if (w < 2) {
        for (int e = lane; e < 2048; e += 32) {
            const int t  = e >> 8;
            const int r  = (e >> 4) & 15;
            const int cc = e & 15;
            const int c  = w * 128 + t * 16 + cc;
            atomicAdd(&fvacc[((size_t)b * ON_ + r) * CN_ + c], red[w * 2048 + e]);
        }
    }
}

// ---------------------------------------------------------------------------
// Kernel 2: MLP head. One block, 256 threads (8 waves x 16 rows = 128 rows).
// GEMM1(+ReLU) -> GEMM2 (prebn out) -> BN stats -> GEMM3 (cls out), all WMMA.
// dynamic LDS: h/prebn buffer [128][256] + mu[256] + rs[256]
// ---------------------------------------------------------------------------
__global__ __launch_bounds__(256) void head_kernel(
    const float* __restrict__ fv,
    const float* __restrict__ W1, const float* __restrict__ b1,
    const float* __restrict__ W2, const float* __restrict__ b2,
    const float* __restrict__ gamma, const float* __restrict__ beta,
    const float* __restrict__ W3, const float* __restrict__ b3,
    const long long* __restrict__ ids,
    float* __restrict__ out, int out_size) {
    extern __shared__ float sm[];
    float* hbuf = sm;                 // [128][256]
    float* muS  = sm + 128 * 256;     // [256]
    float* rsS  = muS + 256;          // [256]

    const int tid  = threadIdx.x;
    const int lane = tid & 31;
    const int w    = tid >> 5;
    const int l15  = lane & 15;
    const int hi2  = (lane >> 4) << 1;
    const int row0 = w * 16;

    v8f acc[16];

    // ---- GEMM1: h = relu(fv/HW @ W1 + b1) -------------------------------
#pragma unroll
    for (int t = 0; t < 16; ++t) acc[t] = zero8();
    {
        const float* arow = fv + (size_t)(row0 + l15) * CN_;
        for (int k = 0; k < CN_; k += 4) {
            v2f a = *(const v2f*)(arow + k + hi2);
            a.x *= INV_HW; a.y *= INV_HW;
#pragma unroll
            for (int t = 0; t < 16; ++t) {
                const float* bp = W1 + (size_t)(k + hi2) * EN_ + t * 16 + l15;
                v2f bb; bb.x = bp[0]; bb.y = bp[EN_];
                acc[t] = wmma4(a, bb, acc[t]);
            }
        }
    }
#pragma unroll
    for (int t = 0; t < 16; ++t)
#pragma unroll
        for (int j = 0; j < 8; ++j) {
            const int r = (lane < 16) ? j : (j + 8);
            float v = acc[t][j] + b1[t * 16 + l15];
            hbuf[(row0 + r) * EN_ + t * 16 + l15] = (v > 0.f) ? v : 0.f;
        }
    __syncthreads();

    // ---- GEMM2: prebn = h @ W2 + b2 (write out + keep in LDS) -----------
#pragma unroll
    for (int t = 0; t < 16; ++t) acc[t] = zero8();
    {
        const float* hrow = hbuf + (size_t)(row0 + l15) * EN_;
        for (int k = 0; k < EN_; k += 4) {
            v2f a = *(const v2f*)(hrow + k + hi2);
#pragma unroll
            for (int t = 0; t < 16; ++t) {
                const float* bp = W2 + (size_t)(k + hi2) * EN_ + t * 16 + l15;
                v2f bb; bb.x = bp[0]; bb.y = bp[EN_];
                acc[t] = wmma4(a, bb, acc[t]);
            }
        }
    }
#pragma unroll
    for (int t = 0; t < 16; ++t)
#pragma unroll
        for (int j = 0; j < 8; ++j) {
            const int r = (lane < 16) ? j : (j + 8);
            const float v = acc[t][j] + b2[t * 16 + l15];
            hbuf[(row0 + r) * EN_ + t * 16 + l15] = v;            // for BN + GEMM3
            out[(size_t)(row0 + r) * EN_ + t * 16 + l15] = v;     // prebn output
        }
    __syncthreads();

    // ---- BatchNorm stats (biased var over 128 rows), one column/thread --
    {
        float s = 0.f, ss = 0.f;
        for (int r = 0; r < 128; ++r) {
            const float x = hbuf[r * EN_ + tid];
            s += x; ss += x * x;
        }
        const float mu  = s * (1.0f / 128.0f);
        const float var = ss * (1.0f / 128.0f) - mu * mu;
        muS[tid] = mu;
        rsS[tid] = rsqrtf(var + 1e-5f);
    }
    __syncthreads();

    // ---- GEMM3: cls = ((prebn-mu)*rs*gamma+beta) @ W3 + b3, N=512 in 2 passes
    float* clsout = out + 128 * EN_;
    for (int p = 0; p < 2; ++p) {
#pragma unroll
        for (int t = 0; t < 16; ++t) acc[t] = zero8();
        const float* hrow = hbuf + (size_t)(row0 + l15) * EN_;
        for (int k = 0; k < EN_; k += 4) {
            const int k0 = k + hi2;
            v2f a;
            a.x = (hrow[k0]     - muS[k0])     * rsS[k0]     * gamma[k0]     + beta[k0];
            a.y = (hrow[k0 + 1] - muS[k0 + 1]) * rsS[k0 + 1] * gamma[k0 + 1] + beta[k0 + 1];
#pragma unroll
            for (int t = 0; t < 16; ++t) {
                const float* bp = W3 + (size_t)k0 * NIN_ + p * 256 + t * 16 + l15;
                v2f bb; bb.x = bp[0]; bb.y = bp[NIN_];
                acc[t] = wmma4(a, bb, acc[t]);
            }
        }
#pragma unroll
        for (int t = 0; t < 16; ++t)
#pragma unroll
            for (int j = 0; j < 8; ++j) {
                const int r = (lane < 16) ? j : (j + 8);
                clsout[(size_t)(row0 + r) * NIN_ + p * 256 + t * 16 + l15] =
                    acc[t][j] + b3[p * 256 + t * 16 + l15];
            }
    }

    // ---- instance ids to output tail ------------------------------------
    if (tid < 128) {
        const int base = 128 * EN_ + 128 * NIN_;
        if (base + tid < out_size) out[base + tid] = (float)ids[tid];
    }
}

// ---------------------------------------------------------------------------
extern "C" void kernel_launch(void* const* d_in, const int* in_sizes, int n_in,
                              void* d_out, int out_size, void* d_ws, size_t ws_size,
                              hipStream_t stream) {
    const float*     features = (const float*)d_in[0];
    const float*     masks    = (const float*)d_in[1];
    const long long* ids      = (const long long*)d_in[2];
    const float*     W1       = (const float*)d_in[3];
    const float*     b1       = (const float*)d_in[4];
    const float*     W2       = (const float*)d_in[5];
    const float*     b2       = (const float*)d_in[6];
    const float*     gamma    = (const float*)d_in[7];
    const float*     beta     = (const float*)d_in[8];
    const float*     W3       = (const float*)d_in[9];
    const float*     b3       = (const float*)d_in[10];
    float* fvacc = (float*)d_ws;                      // [128, 256] f32
    float* out   = (float*)d_out;

    zero_kernel<<<(BN_ * ON_ * CN_ + 255) / 256, 256, 0, stream>>>(fvacc, BN_ * ON_ * CN_);

    pool_kernel<<<dim3(NSLAB, BN_), 256, 0, stream>>>(features, masks, fvacc);

    const size_t shmem = (size_t)(128 * 256 + 2 * 256) * sizeof(float); // 130 KB
    head_kernel<<<1, 256, shmem, stream>>>(fvacc, W1, b1, W2, b2, gamma, beta,
                                           W3, b3, ids, out, out_size);
}